// AdvancedKoopmanModel_75685913690123
// MI455X (gfx1250) — compile-verified
//
#include <hip/hip_runtime.h>

typedef __attribute__((ext_vector_type(2))) float v2f;
typedef __attribute__((ext_vector_type(8))) float v8f;

#define T_NODES 4096
#define D_DIM   128
#define H_DIM   256
#define K_LAT   64
#define U_DIM   4
#define E_EDGES 131072
#define LN_EPS  1e-5f

// ---------------------------------------------------------------------------
// CDNA5 async global->LDS copy (ASYNCcnt-tracked), ISA 15.18.3 op 98.
// vdst = LDS byte address (low 32 bits of the flat shared pointer),
// vaddr = 64-bit global address, GV mode (saddr = off).
// ---------------------------------------------------------------------------
__device__ __forceinline__ void async_ld16(unsigned lds_off, const float* gp)
{
    asm volatile("global_load_async_to_lds_b128 %0, %1, off"
                 :: "v"(lds_off), "v"(gp) : "memory");
}

template <int N>
__device__ __forceinline__ void wait_async()
{
    asm volatile("s_wait_asynccnt %0" :: "i"(N) : "memory");
}

// ---------------------------------------------------------------------------
// GEMM: C[M,N] = act(A[M,Kd] @ W[Kd,N] + bias)
// f32 WMMA 16x16x4. Block = 128 threads (4 waves), tile 64(M) x 64(N),
// each wave owns a 32x32 sub-tile (2x2 WMMA tiles).
// K staged 16-wide into double-buffered LDS via async global->LDS copies;
// fragments for the whole chunk are batch-loaded, then 16 WMMAs burst.
// Requires: M % 64 == 0, N % 64 == 0, Kd % 16 == 0 (true for all calls here).
// ---------------------------------------------------------------------------
__global__ __launch_bounds__(128)
void gemm_wmma_f32(const float* __restrict__ A, const float* __restrict__ W,
                   const float* __restrict__ bias, float* __restrict__ C,
                   int M, int Kd, int N, int act)
{
    // stride 20 floats = 80B: 16B-aligned rows (async B128) and
    // (20*m + k) % 64 hits 16 distinct banks for m = 0..15.
    __shared__ __align__(16) float As[2][64][20];
    // stride 72 floats = 288B: 16B-aligned; lane groups (rows r, r+2) map to
    // disjoint bank ranges (8r+n vs 8r+16+n).
    __shared__ __align__(16) float Bs[2][16][72];

    const int tid  = threadIdx.x;
    const int wave = tid >> 5;          // 0..3
    const int lane = tid & 31;
    const int wm   = wave >> 1;         // wave row (0..1)
    const int wn   = wave & 1;          // wave col (0..1)
    const int lm   = lane & 15;
    const int lg   = lane >> 4;         // lane group (0..1)

    const int blockM = blockIdx.x * 64;
    const int blockN = blockIdx.y * 64;

    // per-thread staging slots: 2 float4 of A panel + 2 float4 of B panel
    int a_row[2], a_col[2], b_row[2], b_col[2];
    const float* a_gp[2];
    const float* b_gp[2];
    #pragma unroll
    for (int r = 0; r < 2; ++r) {
        int idx  = tid + r * 128;              // 0..255
        a_row[r] = idx >> 2;                   // 0..63
        a_col[r] = (idx & 3) * 4;              // 0,4,8,12
        b_row[r] = idx >> 4;                   // 0..15
        b_col[r] = (idx & 15) * 4;             // 0..60
        a_gp[r]  = A + (size_t)(blockM + a_row[r]) * Kd + a_col[r];
        b_gp[r]  = W + (size_t)b_row[r] * N + blockN + b_col[r];
    }

    auto stage = [&](int b, int c) {
        #pragma unroll
        for (int r = 0; r < 2; ++r)
            async_ld16((unsigned)(size_t)&As[b][a_row[r]][a_col[r]],
                       a_gp[r] + (size_t)c * 16);
        #pragma unroll
        for (int r = 0; r < 2; ++r)
            async_ld16((unsigned)(size_t)&Bs[b][b_row[r]][b_col[r]],
                       b_gp[r] + (size_t)c * 16 * N);
    };

    v8f acc[2][2];
    #pragma unroll
    for (int i = 0; i < 2; ++i)
        #pragma unroll
        for (int j = 0; j < 2; ++j) acc[i][j] = {};

    const int nch = Kd >> 4;
    stage(0, 0);                                    // prologue: chunk 0 in flight

    for (int c = 0; c < nch; ++c) {
        const int cur = c & 1;
        if (c + 1 < nch) { stage(cur ^ 1, c + 1);   // prefetch next chunk
                           wait_async<4>(); }       // chunk c landed (in-order)
        else             { wait_async<0>(); }
        __syncthreads();

        const float (*Asb)[20] = As[cur];
        const float (*Bsb)[72] = Bs[cur];

        // batch-load all fragments for the 16-wide chunk, then WMMA burst
        v2f af[4][2], bf[4][2];
        #pragma unroll
        for (int ks = 0; ks < 4; ++ks) {
            int kb = ks * 4 + lg * 2;   // ISA: VGPR v holds K = v + 2*laneGroup
            #pragma unroll
            for (int i = 0; i < 2; ++i) {
                int m = wm * 32 + i * 16 + lm;
                af[ks][i].x = Asb[m][kb];
                af[ks][i].y = Asb[m][kb + 1];
            }
            #pragma unroll
            for (int j = 0; j < 2; ++j) {
                int n = wn * 32 + j * 16 + lm;
                bf[ks][j].x = Bsb[kb][n];
                bf[ks][j].y = Bsb[kb + 1][n];
            }
        }
        #pragma unroll
        for (int ks = 0; ks < 4; ++ks)
            #pragma unroll
            for (int i = 0; i < 2; ++i)
                #pragma unroll
                for (int j = 0; j < 2; ++j)
                    acc[i][j] = __builtin_amdgcn_wmma_f32_16x16x4_f32(
                        false, af[ks][i], false, bf[ks][j],
                        (short)0, acc[i][j], false, false);
        __syncthreads();   // protect buffer reused by the prefetch 2 chunks out
    }

    // --- epilogue: bias + optional ReLU; C/D layout M = r + 8*laneGroup ---
    #pragma unroll
    for (int i = 0; i < 2; ++i) {
        #pragma unroll
        for (int j = 0; j < 2; ++j) {
            int nn = blockN + wn * 32 + j * 16 + lm;
            float bv = bias ? bias[nn] : 0.0f;
            #pragma unroll
            for (int r = 0; r < 8; ++r) {
                int mm = blockM + wm * 32 + i * 16 + lg * 8 + r;
                float v = acc[i][j][r] + bv;
                if (act) v = fmaxf(v, 0.0f);
                C[(size_t)mm * N + nn] = v;
            }
        }
    }
}

// ---------------------------------------------------------------------------
// Degree / normalization
// ---------------------------------------------------------------------------
__global__ void deg_count_kernel(const int* __restrict__ dst, float* __restrict__ deg, int E)
{
    int i = blockIdx.x * blockDim.x + threadIdx.x;
    if (i < E) atomicAdd(&deg[dst[i]], 1.0f);
}

__global__ void dinv_kernel(const float* __restrict__ deg, float* __restrict__ dinv, int n)
{
    int i = blockIdx.x * blockDim.x + threadIdx.x;
    if (i < n) dinv[i] = rsqrtf(1.0f + deg[i]);
}

// out[t,j] = hw[t,j]*dinv[t]^2 + b[j]   (self-loop term + bias)
__global__ void gcn_init_kernel(const float* __restrict__ hw, const float* __restrict__ bias,
                                const float* __restrict__ dinv, float* __restrict__ out,
                                int n, int dim)
{
    int i = blockIdx.x * blockDim.x + threadIdx.x;
    if (i < n * dim) {
        int t = i / dim, j = i - t * dim;
        float dn = dinv[t];
        out[i] = hw[i] * dn * dn + bias[j];
    }
}

// out[dst[e], :] += dinv[src]*dinv[dst] * hw[src[e], :]   (4 features / thread)
__global__ void gcn_scatter_kernel(const float* __restrict__ hw,
                                   const int* __restrict__ src, const int* __restrict__ dst,
                                   const float* __restrict__ dinv, float* __restrict__ out,
                                   int dim4, int dim, int E)
{
    int idx = blockIdx.x * blockDim.x + threadIdx.x;
    int e = idx / dim4;
    if (e >= E) return;
    int c = (idx - e * dim4) * 4;
    int s = src[e], d = dst[e];
    float ne = dinv[s] * dinv[d];
    float4 v = *(const float4*)(hw + (size_t)s * dim + c);
    float* o = out + (size_t)d * dim + c;
    atomicAdd(o + 0, ne * v.x);
    atomicAdd(o + 1, ne * v.y);
    atomicAdd(o + 2, ne * v.z);
    atomicAdd(o + 3, ne * v.w);
}

// in-place h = layernorm(relu(h)) * g + b ; one wave32 per row, dim in {64,128,256}
__global__ __launch_bounds__(256)
void relu_layernorm_kernel(float* __restrict__ buf, const float* __restrict__ g,
                           const float* __restrict__ b, int n, int dim)
{
    int row  = blockIdx.x * 8 + (threadIdx.x >> 5);
    int lane = threadIdx.x & 31;
    if (row >= n) return;
    float* p  = buf + (size_t)row * dim;
    int   per = dim >> 5;                 // 2, 4, or 8
    float vals[8];
    float sum = 0.0f;
    #pragma unroll
    for (int i = 0; i < 8; ++i)
        if (i < per) { float v = fmaxf(p[lane + i * 32], 0.0f); vals[i] = v; sum += v; }
    #pragma unroll
    for (int off = 16; off > 0; off >>= 1) sum += __shfl_xor(sum, off, 32);
    float mu = sum / (float)dim;
    float var = 0.0f;
    #pragma unroll
    for (int i = 0; i < 8; ++i)
        if (i < per) { float dlt = vals[i] - mu; var += dlt * dlt; }
    #pragma unroll
    for (int off = 16; off > 0; off >>= 1) var += __shfl_xor(var, off, 32);
    float inv = rsqrtf(var / (float)dim + LN_EPS);
    #pragma unroll
    for (int i = 0; i < 8; ++i)
        if (i < per) { int c = lane + i * 32; p[c] = (vals[i] - mu) * inv * g[c] + b[c]; }
}

__global__ void combine_kernel(const float* __restrict__ a, const float* __restrict__ bb,
                               float* __restrict__ out, int n)
{
    int i = blockIdx.x * blockDim.x + threadIdx.x;
    if (i < n) out[i] = (a[i] + bb[i]) * 0.5f;
}

// u_proj[t, j] = sum_u edge_attr[t,u] * Lw[j,u]   (Kd=4: scalar kernel)
__global__ void uproj_kernel(const float* __restrict__ ea, const float* __restrict__ Lw,
                             float* __restrict__ up, int n)
{
    int idx = blockIdx.x * blockDim.x + threadIdx.x;
    int t = idx >> 6, j = idx & 63;
    if (t >= n) return;
    float4 e = *(const float4*)(ea + (size_t)t * 4);
    float4 l = *(const float4*)(Lw + (size_t)j * 4);
    up[(size_t)t * 64 + j] = e.x * l.x + e.y * l.y + e.z * l.z + e.w * l.w;
}

// Sequential Koopman rollout: ghat[0]=koop[0]; ghat[t]=ghat[t-1]@A + up[t-1]
// One 64-thread block; thread j caches column A[:,j] in registers.
__global__ __launch_bounds__(64)
void koopman_scan_kernel(const float* __restrict__ koop, const float* __restrict__ Amat,
                         const float* __restrict__ up, float* __restrict__ ghat, int Tn)
{
    __shared__ float gprev[64];
    int j = threadIdx.x;
    float colA[64];
    #pragma unroll
    for (int i = 0; i < 64; ++i) colA[i] = Amat[i * 64 + j];
    float g0 = koop[j];
    gprev[j] = g0;
    ghat[j]  = g0;
    __syncthreads();
    for (int t = 1; t < Tn; ++t) {
        float s = up[(size_t)(t - 1) * 64 + j];
        #pragma unroll
        for (int i = 0; i < 64; ++i) s += gprev[i] * colA[i];
        __syncthreads();
        gprev[j] = s;
        ghat[(size_t)t * 64 + j] = s;
        __syncthreads();
    }
}

// ---------------------------------------------------------------------------
// Host-side orchestration
// ---------------------------------------------------------------------------
struct GnnP {
    const float *fc1w,*fc1b,*fc2w,*fc2b,*fc3w,*fc3b,*fc4w,*fc4b;
    const float *c1w,*c1b,*c2w,*c2b,*c3w,*c3b;
    const float *n1g,*n1b,*n2g,*n2b,*n3g,*n3b;
};

static GnnP load_params(void* const* d_in, const int* in_sizes, int base)
{
    auto f = [&](int o) { return (const float*)d_in[base + o]; };
    GnnP p;
    // insertion order: fc2.w (H*H=65536) sits at offset 2; alphabetical pytree
    // order puts conv2.w (H*H/2=32768) there instead.
    if (in_sizes[base + 2] == H_DIM * H_DIM) {
        p.fc1w=f(0);  p.fc1b=f(1);  p.fc2w=f(2);  p.fc2b=f(3);
        p.fc3w=f(4);  p.fc3b=f(5);  p.fc4w=f(6);  p.fc4b=f(7);
        p.c1w =f(8);  p.c1b =f(9);  p.c2w =f(10); p.c2b =f(11);
        p.c3w =f(12); p.c3b =f(13);
        p.n1g =f(14); p.n1b =f(15); p.n2g =f(16); p.n2b =f(17);
        p.n3g =f(18); p.n3b =f(19);
    } else {
        p.c1w =f(0);  p.c1b =f(1);  p.c2w =f(2);  p.c2b =f(3);
        p.c3w =f(4);  p.c3b =f(5);
        p.fc1w=f(6);  p.fc1b=f(7);  p.fc2w=f(8);  p.fc2b=f(9);
        p.fc3w=f(10); p.fc3b=f(11); p.fc4w=f(12); p.fc4b=f(13);
        p.n1g =f(14); p.n1b =f(15); p.n2g =f(16); p.n2b =f(17);
        p.n3g =f(18); p.n3b =f(19);
    }
    return p;
}

static void launch_gemm(const float* A, const float* W, const float* bias, float* C,
                        int M, int Kd, int N, int act, hipStream_t stream)
{
    dim3 grid(M / 64, N / 64), block(128);
    hipLaunchKernelGGL(gemm_wmma_f32, grid, block, 0, stream, A, W, bias, C, M, Kd, N, act);
}

static void run_gnn(const float* input, int din, int dout, const GnnP& p, float* dest,
                    const int* esrc, const int* edst, const float* dinv,
                    float* hw, float* bufA, float* bufB, float* fA, float* fB,
                    hipStream_t stream)
{
    const int Tn = T_NODES, H = H_DIM, H2 = H_DIM / 2;
    auto gcn = [&](const float* in, int k, int n, const float* w, const float* b,
                   const float* ng, const float* nb, float* out) {
        launch_gemm(in, w, nullptr, hw, Tn, k, n, 0, stream);
        hipLaunchKernelGGL(gcn_init_kernel, dim3((Tn * n + 255) / 256), dim3(256), 0, stream,
                           hw, b, dinv, out, Tn, n);
        int d4 = n / 4;
        hipLaunchKernelGGL(gcn_scatter_kernel, dim3((E_EDGES * d4 + 255) / 256), dim3(256), 0, stream,
                           hw, esrc, edst, dinv, out, d4, n, E_EDGES);
        hipLaunchKernelGGL(relu_layernorm_kernel, dim3(Tn / 8), dim3(256), 0, stream,
                           out, ng, nb, Tn, n);
    };
    gcn(input, din, H,   p.c1w, p.c1b, p.n1g, p.n1b, bufA);   // g1 [T,256]
    gcn(bufA,  H,   H2,  p.c2w, p.c2b, p.n2g, p.n2b, bufB);   // g2 [T,128]
    gcn(bufB,  H2,  dout,p.c3w, p.c3b, p.n3g, p.n3b, bufA);   // g3 [T,dout]

    launch_gemm(input, p.fc1w, p.fc1b, fA, Tn, din, H, 1, stream);
    launch_gemm(fA,    p.fc2w, p.fc2b, fB, Tn, H,   H, 1, stream);
    launch_gemm(fB,    p.fc3w, p.fc3b, fA, Tn, H,  H2, 1, stream);
    launch_gemm(fA,    p.fc4w, p.fc4b, fB, Tn, H2, dout, 0, stream);

    hipLaunchKernelGGL(combine_kernel, dim3((Tn * dout + 255) / 256), dim3(256), 0, stream,
                       bufA, fB, dest, Tn * dout);
}

extern "C" void kernel_launch(void* const* d_in, const int* in_sizes, int n_in,
                              void* d_out, int out_size, void* d_ws, size_t ws_size,
                              hipStream_t stream)
{
    const float* x    = (const float*)d_in[0];
    const int*   esrc = (const int*)d_in[1];
    const int*   edst = (const int*)d_in[2];
    const float* eatt = (const float*)d_in[3];
    GnnP enc = load_params(d_in, in_sizes, 4);
    GnnP dec = load_params(d_in, in_sizes, 24);
    const float* Amat = (const float*)d_in[44];
    const float* Lw   = (const float*)d_in[45];

    float* out = (float*)d_out;
    float* decoded_ae      = out;                                   // [T, D]
    float* decoded_rollout = out + (size_t)T_NODES * D_DIM;         // [T, D]
    float* koop_out        = out + (size_t)2 * T_NODES * D_DIM;     // [T, K]

    // workspace layout (floats)
    float* w      = (float*)d_ws;
    float* deg    = w;                         w += T_NODES;
    float* dinv   = w;                         w += T_NODES;
    float* hw     = w;                         w += (size_t)T_NODES * H_DIM;
    float* bufA   = w;                         w += (size_t)T_NODES * H_DIM;
    float* bufB   = w;                         w += (size_t)T_NODES * H_DIM;
    float* fA     = w;                         w += (size_t)T_NODES * H_DIM;
    float* fB     = w;                         w += (size_t)T_NODES * H_DIM;
    float* koop   = w;                         w += (size_t)T_NODES * K_LAT;
    float* ghat   = w;                         w += (size_t)T_NODES * K_LAT;
    float* uproj  = w;                         w += (size_t)T_NODES * K_LAT;

    // --- GCN normalization ---
    hipMemsetAsync(deg, 0, T_NODES * sizeof(float), stream);
    hipLaunchKernelGGL(deg_count_kernel, dim3((E_EDGES + 255) / 256), dim3(256), 0, stream,
                       edst, deg, E_EDGES);
    hipLaunchKernelGGL(dinv_kernel, dim3((T_NODES + 255) / 256), dim3(256), 0, stream,
                       deg, dinv, T_NODES);

    // --- encoder: koop = gnn(x) ---
    run_gnn(x, D_DIM, K_LAT, enc, koop, esrc, edst, dinv, hw, bufA, bufB, fA, fB, stream);
    hipMemcpyAsync(koop_out, koop, (size_t)T_NODES * K_LAT * sizeof(float),
                   hipMemcpyDeviceToDevice, stream);

    // --- decoded_ae = gnn(koop) ---
    run_gnn(koop, K_LAT, D_DIM, dec, decoded_ae, esrc, edst, dinv, hw, bufA, bufB, fA, fB, stream);

    // --- Koopman rollout ---
    hipLaunchKernelGGL(uproj_kernel, dim3(((T_NODES - 1) * K_LAT + 255) / 256), dim3(256), 0, stream,
                       eatt, Lw, uproj, T_NODES - 1);
    hipLaunchKernelGGL(koopman_scan_kernel, dim3(1), dim3(64), 0, stream,
                       koop, Amat, uproj, ghat, T_NODES);

    // --- decoded_rollout = gnn(ghat) ---
    run_gnn(ghat, K_LAT, D_DIM, dec, decoded_rollout, esrc, edst, dinv, hw, bufA, bufB, fA, fB, stream);
}